// GaussianFalloffKernel_14688788152367
// MI455X (gfx1250) — compile-verified
//
#include <hip/hip_runtime.h>

// Elementwise vortex-velocity kernel for MI455X (gfx1250, wave32).
// Bandwidth-bound: 256 MiB total traffic -> ~11 us floor @ 23.3 TB/s.
// Strategy: b128 non-temporal loads/stores, wave-uniform scalar vortex
// params (s_load), hardware exp2/rsq/rcp transcendentals.

typedef __attribute__((ext_vector_type(4))) float v4f;

#define THREADS 256
#define UNROLL  2   // float4s per thread, block-strided

__device__ __forceinline__ v4f vortex2(v4f a, float y, float x, float tau, float c) {
    // a = {p0.c0, p0.c1, p1.c0, p1.c1}
    float d1a = a.x - y, d2a = a.y - x;
    float d1b = a.z - y, d2b = a.w - x;
    float r2a = d1a * d1a + d2a * d2a;
    float r2b = d1b * d1b + d2b * d2b;
    // exp(-r2/sigma^2)/sqrt(r2) = exp2(c*r2) * rsq(r2), c = -log2(e)/sigma^2
    float sa = tau * __builtin_amdgcn_exp2f(c * r2a) * __builtin_amdgcn_rsqf(r2a);
    float sb = tau * __builtin_amdgcn_exp2f(c * r2b) * __builtin_amdgcn_rsqf(r2b);
    v4f r;
    r.x =  sa * d2a;
    r.y = -sa * d1a;
    r.z =  sb * d2b;
    r.w = -sb * d1b;
    return r;
}

__global__ __launch_bounds__(THREADS) void vortex_velocity_kernel(
    const float* __restrict__ vf,   // (B,1,6): y,x,tau,sigma,...
    const v4f*  __restrict__ pts,   // (B,H,W,2) viewed as float4
    v4f*        __restrict__ out,   // (B,H,W,2) viewed as float4
    int f4_per_b)                   // float4 elements per batch
{
    const int b = blockIdx.y;       // wave-uniform -> scalar loads below
    const float y   = vf[b * 6 + 0];
    const float x   = vf[b * 6 + 1];
    const float tau = vf[b * 6 + 2];
    const float sig = vf[b * 6 + 3];
    // hardware reciprocal instead of IEEE division ladder
    const float c = -1.4426950408889634f * __builtin_amdgcn_rcpf(sig * sig);

    const v4f* __restrict__ p = pts + (size_t)b * f4_per_b;
    v4f* __restrict__ o       = out + (size_t)b * f4_per_b;

    const int i0 = blockIdx.x * (THREADS * UNROLL) + threadIdx.x;
    const int i1 = i0 + THREADS;

    // Full tiles (the only path taken for the reference shape: 131072 % 512 == 0).
    if (i1 < f4_per_b) {
        v4f a0 = __builtin_nontemporal_load(p + i0);
        v4f a1 = __builtin_nontemporal_load(p + i1);
        v4f r0 = vortex2(a0, y, x, tau, c);
        v4f r1 = vortex2(a1, y, x, tau, c);
        __builtin_nontemporal_store(r0, o + i0);
        __builtin_nontemporal_store(r1, o + i1);
    } else if (i0 < f4_per_b) {
        v4f a0 = __builtin_nontemporal_load(p + i0);
        v4f r0 = vortex2(a0, y, x, tau, c);
        __builtin_nontemporal_store(r0, o + i0);
    }
}

extern "C" void kernel_launch(void* const* d_in, const int* in_sizes, int n_in,
                              void* d_out, int out_size, void* d_ws, size_t ws_size,
                              hipStream_t stream) {
    const float* vf  = (const float*)d_in[0];   // (B,1,6) float32
    const v4f*   pts = (const v4f*)d_in[1];     // (B,H,W,2) float32
    v4f*         out = (v4f*)d_out;             // (B,H,W,2) float32

    const int B = in_sizes[0] / 6;              // 64
    const int floats_per_b = in_sizes[1] / B;   // H*W*2 = 524288
    const int f4_per_b = floats_per_b / 4;      // 131072

    const int xblocks = (f4_per_b + THREADS * UNROLL - 1) / (THREADS * UNROLL); // 256
    dim3 grid(xblocks, B);
    dim3 block(THREADS);
    vortex_velocity_kernel<<<grid, block, 0, stream>>>(vf, pts, out, f4_per_b);
}